// LockedMemoryNeuron_44976897523913
// MI455X (gfx1250) — compile-verified
//
#include <hip/hip_runtime.h>
#include <hip/hip_bf16.h>

typedef float v2f __attribute__((ext_vector_type(2)));
typedef float v8f __attribute__((ext_vector_type(8)));

#define NROWS   (16 * 2048)          // B*T
#define DMODEL  1024
#define DSTATE  256
#define G3      (3 * DSTATE)         // 768
#define OUT_ELEMS ((size_t)NROWS * DMODEL)

static constexpr int LDT = 20;       // padded LDS row stride: 80B (16B aligned),
                                     // 20*l mod 64 distinct for l=0..15

// ---- CDNA5 async global->LDS staging -------------------------------------
// Per-lane 16B copy straight into LDS (no VGPR round trip), tracked on
// ASYNCcnt. LDS operand = low 32 bits of the generic pointer (flat->LDS
// mapping is addr[31:0] per the aperture rules).
__device__ __forceinline__ void async_ld_b128(float* lds, const float* gsrc) {
  uint32_t l = (uint32_t)(uintptr_t)lds;
  asm volatile("global_load_async_to_lds_b128 %0, %1, off"
               :: "v"(l), "v"(gsrc) : "memory");
}
__device__ __forceinline__ void wait_async_all() {
#if __has_builtin(__builtin_amdgcn_s_wait_asynccnt)
  __builtin_amdgcn_s_wait_asynccnt(0);
#else
  asm volatile("s_wait_asynccnt 0x0" ::: "memory");
#endif
}

// ---- fast transcendentals for the scan critical path ---------------------
__device__ __forceinline__ float fast_sigmoid(float x) {
  // 1/(1+exp(-x)) via v_exp_f32 (base-2) + v_rcp_f32
  return __builtin_amdgcn_rcpf(
      1.0f + __builtin_amdgcn_exp2f(x * -1.4426950408889634f));
}
__device__ __forceinline__ float fast_tanh(float x) {
#if __has_builtin(__builtin_amdgcn_tanhf)
  return __builtin_amdgcn_tanhf(x);
#elif __has_builtin(__builtin_amdgcn_tanh_f32)
  return __builtin_amdgcn_tanh_f32(x);
#else
  float y;
  // TRANS op: result needs 1 independent op before use (ISA §7.4); pad nops.
  asm volatile("v_tanh_f32 %0, %1\n\tv_nop\n\tv_nop" : "=v"(y) : "v"(x));
  return y;
#endif
}

// ---------------------------------------------------------------------------
// C[N,P] = A[N,K] @ W[P,K]^T + bias[P]
// Block: 256 threads (8 waves), tile 64(M) x 128(P). K staged 16 at a time
// into double-buffered LDS via async-to-LDS; next chunk's loads are issued
// before computing the current chunk (one barrier per chunk). Each wave owns
// a 32x32 patch = 2x2 accumulators of V_WMMA_F32_16X16X4_F32.
// ---------------------------------------------------------------------------
__global__ __launch_bounds__(256) void gemm_nt_wmma(
    const float* __restrict__ A,     // [N,K] row-major
    const float* __restrict__ W,     // [P,K] row-major
    const float* __restrict__ bias,  // [P]
    float* __restrict__ C,           // [N,P]
    int N, int K, int P)
{
  __shared__ float lA[2 * 64 * LDT];
  __shared__ float lW[2 * 128 * LDT];

  const int tid  = threadIdx.x;
  const int wid  = tid >> 5;
  const int lane = tid & 31;
  const int l    = lane & 15;
  const int kh   = (lane >> 4) << 1;     // lanes 0-15 -> K+0/1, 16-31 -> K+2/3

  const int pBlocks = P >> 7;            // P / 128
  const int bm   = blockIdx.x / pBlocks;
  const int bp   = blockIdx.x - bm * pBlocks;
  const int row0 = bm << 6;
  const int col0 = bp << 7;

  const int wm = (wid >> 2) << 5;        // 0 / 32 within block tile (M)
  const int wp = (wid & 3)  << 5;        // 0..96 within block tile (P)

  v8f acc00 = {}, acc01 = {}, acc10 = {}, acc11 = {};

  auto stage = [&](int kc, int buf) {
    float* dA = lA + buf * (64 * LDT);
    float* dW = lW + buf * (128 * LDT);
    {   // A tile 64x16: one b128 per thread
      const int r = tid >> 2, s = (tid & 3) << 2;
      async_ld_b128(&dA[r * LDT + s], &A[(size_t)(row0 + r) * K + kc + s]);
    }
#pragma unroll
    for (int i = 0; i < 2; ++i) {   // W tile 128x16: two b128 per thread
      const int idx = tid + (i << 8);
      const int r = idx >> 2, s = (idx & 3) << 2;
      async_ld_b128(&dW[r * LDT + s], &W[(size_t)(col0 + r) * K + kc + s]);
    }
  };

  stage(0, 0);
  wait_async_all();
  __syncthreads();

  int buf = 0;
  for (int kc = 0; kc < K; kc += 16) {
    if (kc + 16 < K) stage(kc + 16, buf ^ 1);   // prefetch next chunk

    const float* bA = lA + buf * (64 * LDT);
    const float* bW = lW + buf * (128 * LDT);
#pragma unroll
    for (int kk = 0; kk < 16; kk += 4) {
      const int ko = kk + kh;
      // A fragment (16x4 fp32): lane holds [M=l, K=ko], [M=l, K=ko+1]
      const float* pa0 = &bA[(wm +      l) * LDT + ko];
      const float* pa1 = &bA[(wm + 16 + l) * LDT + ko];
      // B fragment (4x16 fp32): lane holds [K=ko, N=l], [K=ko+1, N=l]
      const float* pb0 = &bW[(wp +      l) * LDT + ko];
      const float* pb1 = &bW[(wp + 16 + l) * LDT + ko];
      v2f a0; a0.x = pa0[0]; a0.y = pa0[1];
      v2f a1; a1.x = pa1[0]; a1.y = pa1[1];
      v2f b0; b0.x = pb0[0]; b0.y = pb0[1];
      v2f b1; b1.x = pb1[0]; b1.y = pb1[1];
      acc00 = __builtin_amdgcn_wmma_f32_16x16x4_f32(false, a0, false, b0,
                                                    (short)0, acc00, false, false);
      acc01 = __builtin_amdgcn_wmma_f32_16x16x4_f32(false, a0, false, b1,
                                                    (short)0, acc01, false, false);
      acc10 = __builtin_amdgcn_wmma_f32_16x16x4_f32(false, a1, false, b0,
                                                    (short)0, acc10, false, false);
      acc11 = __builtin_amdgcn_wmma_f32_16x16x4_f32(false, a1, false, b1,
                                                    (short)0, acc11, false, false);
    }

    wait_async_all();   // this wave's prefetch landed in LDS
    __syncthreads();    // ... and everyone else's too
    buf ^= 1;
  }

  // C/D layout: VGPR r -> M = r (lanes 0-15) / M = r+8 (lanes 16-31), N = l
  const int mhi = (lane >> 4) << 3;
  const int c0  = col0 + wp + l;
  const int c1  = c0 + 16;
  const float bb0 = bias[c0];
  const float bb1 = bias[c1];
  const int m0 = row0 + wm + mhi;
  const int m1 = m0 + 16;
#pragma unroll
  for (int r = 0; r < 8; ++r) {
    C[(size_t)(m0 + r) * P + c0] = acc00[r] + bb0;
    C[(size_t)(m0 + r) * P + c1] = acc01[r] + bb1;
    C[(size_t)(m1 + r) * P + c0] = acc10[r] + bb0;
    C[(size_t)(m1 + r) * P + c1] = acc11[r] + bb1;
  }
}

// ---------------------------------------------------------------------------
// Locked-GRU scan: recurrence is elementwise per (b,d) since W_hr=W_hz=0 and
// W_hn=I. 16 blocks x 256 threads = 4096 independent scalar chains. The
// dependent chain per step is exp2/rcp (r,z), v_tanh (n), two fmas; gx loads
// are h-independent so unroll-4 lets them pipeline ahead.
// ---------------------------------------------------------------------------
__global__ __launch_bounds__(256) void gru_scan(
    const float* __restrict__ gx,    // [B,T,768]
    const float* __restrict__ h_in,  // [1,B,256]
    float* __restrict__ o,           // [B,T,256]
    float* __restrict__ hlast,       // [B,256]
    int T)
{
  const int b = blockIdx.x;
  const int d = threadIdx.x;
  float h = h_in[b * DSTATE + d];
  const float* g = gx + (size_t)b * T * G3 + d;
  float*      op = o  + (size_t)b * T * DSTATE + d;
#pragma unroll 4
  for (int t = 0; t < T; ++t) {
    const float gr = g[0];
    const float gz = g[DSTATE];
    const float gn = g[2 * DSTATE];
    g += G3;
    const float r = fast_sigmoid(gr);
    const float z = fast_sigmoid(gz + 5.0f);
    const float n = fast_tanh(fmaf(r, h, gn));
    h = fmaf(z, h - n, n);           // (1-z)*n + z*h
    *op = h;
    op += DSTATE;
  }
  hlast[b * DSTATE + d] = h;
}

// ---------------------------------------------------------------------------
// Staging plan (all stream-ordered, graph-capture safe):
//   1) gx = x @ W_ih^T + b_ih   -> stored in d_out[0 .. 25.1M floats)
//      (d_out holds 33.55M floats; gx fits and does not touch the h_last
//       tail at element 33,554,432)
//   2) scan: reads gx, writes o -> d_ws (needs 32 MiB), h_last -> d_out tail
//   3) out = o @ W_out^T + b_out -> overwrites d_out[0 .. 33.55M) (gx dead)
// ---------------------------------------------------------------------------
extern "C" void kernel_launch(void* const* d_in, const int* in_sizes, int n_in,
                              void* d_out, int out_size, void* d_ws, size_t ws_size,
                              hipStream_t stream) {
  const float* x     = (const float*)d_in[0];
  const float* h0    = (const float*)d_in[1];
  const float* W_ih  = (const float*)d_in[2];
  const float* b_ih  = (const float*)d_in[3];
  const float* W_out = (const float*)d_in[4];
  const float* b_out = (const float*)d_in[5];

  float* out   = (float*)d_out;
  float* gx    = out;                    // scratch inside d_out
  float* o     = (float*)d_ws;           // NROWS*DSTATE*4 = 32 MiB
  float* hlast = out + OUT_ELEMS;

  dim3 blk(256);
  // gx = x @ W_ih^T + b_ih : N=32768, K=1024, P=768 -> 512*6 blocks
  gemm_nt_wmma<<<dim3((NROWS / 64) * (G3 / 128)), blk, 0, stream>>>(
      x, W_ih, b_ih, gx, NROWS, DMODEL, G3);
  // sequential scan over T
  gru_scan<<<dim3(16), blk, 0, stream>>>(gx, h0, o, hlast, 2048);
  // out = o @ W_out^T + b_out : N=32768, K=256, P=1024 -> 512*8 blocks
  gemm_nt_wmma<<<dim3((NROWS / 64) * (DMODEL / 128)), blk, 0, stream>>>(
      o, W_out, b_out, out, NROWS, DSTATE, DMODEL);
}